// PositionalEncoding2D_68693706932425
// MI455X (gfx1250) — compile-verified
//
#include <hip/hip_runtime.h>
#include <stdint.h>

// Problem constants (from reference)
#define NSAMP   65536
#define EMB     1536
#define BINS    257
#define PDIM    (BINS * BINS)      // 66049
#define ROW4    (EMB / 4)          // 384 float4 per row
#define SPB     32                 // samples per block in gather kernels

// Native clang vector (legal for __builtin_nontemporal_*; HIP float4 is not)
typedef float v4f __attribute__((ext_vector_type(4)));

// ---------------------------------------------------------------------------
// gfx1250 async global->LDS copy (ASYNCcnt path) + wait.
// pe is streamed exactly once -> non-temporal hint keeps L2 for the hot
// transposed-table rows used by the gather phase.
// ---------------------------------------------------------------------------
__device__ __forceinline__ void async_g2lds_b32_nt(uint32_t lds_addr, const float* gaddr) {
#if defined(__gfx1250__)
    asm volatile("global_load_async_to_lds_b32 %0, %1, off th:TH_LOAD_NT"
                 :: "v"(lds_addr), "v"(gaddr)
                 : "memory");
#else
    (void)lds_addr; (void)gaddr;
#endif
}

__device__ __forceinline__ void wait_asynccnt0() {
#if defined(__gfx1250__)
    asm volatile("s_wait_asynccnt 0x0" ::: "memory");
#endif
}

// Low 32 bits of a generic pointer into LDS == LDS byte offset on AMDGPU.
__device__ __forceinline__ uint32_t lds_addr_of(const void* p) {
    return (uint32_t)(uintptr_t)p;
}

// ---------------------------------------------------------------------------
// Bit-exact searchsorted(linspace(-5,5,257), x, side='right') after clip.
// Edges -5 + k*(10/256) are exactly representable in f32, so a floor estimate
// refined against exact edges matches the reference exactly.
// ---------------------------------------------------------------------------
__device__ __forceinline__ float edgef(int i) {
    // avoid fma contraction: two explicitly rounded ops (both exact here)
    return __fadd_rn(__fmul_rn((float)i, 0.0390625f), -5.0f);
}

__device__ __forceinline__ int bin_index(float x) {
    const float MN = -5.0f;
    const float MXE = 5.0f - 1e-6f;         // MAX - EPS (clip upper bound)
    x = fminf(fmaxf(x, MN), MXE);
    int k = (int)floorf(__fmul_rn(__fsub_rn(x, MN), 25.6f));
    k = min(max(k, 0), 255);
    if (x >= edgef(k + 1)) k += 1;          // estimate one too low
    if (x <  edgef(k))     k -= 1;          // estimate one too high
    return k + 1;                            // side='right'
}

// ---------------------------------------------------------------------------
// Kernel 1: transpose pe[EMB][PDIM] -> pet[PDIM][EMB] (32x32 LDS tiles).
// Loads staged directly into LDS with gfx1250 async copies (NT hint).
// grid = (ceil(PDIM/32), EMB/32), block = (32, 8)
// ---------------------------------------------------------------------------
__global__ __launch_bounds__(256) void pe_transpose_kernel(
        const float* __restrict__ pe, float* __restrict__ pet) {
    __shared__ float tile[32][33];

    const int p0 = blockIdx.x * 32;
    const int e0 = blockIdx.y * 32;
    const int tx = threadIdx.x;   // 0..31
    const int ty = threadIdx.y;   // 0..7

    // Load phase: rows e = e0+ty+8k, col p = p0+tx (coalesced along p).
    // Clamp p for the single ragged tile (PDIM % 32 == 1); the store phase
    // bounds-check discards the duplicated data.
    const int pclamped = min(p0 + tx, PDIM - 1);
#pragma unroll
    for (int k = 0; k < 4; ++k) {
        const int e = e0 + ty + 8 * k;   // always < EMB (grid exact in e)
        async_g2lds_b32_nt(lds_addr_of(&tile[ty + 8 * k][tx]),
                           pe + (size_t)e * PDIM + pclamped);
    }
    wait_asynccnt0();
    __syncthreads();

    // Store phase: pet[p][e], e contiguous across lanes (coalesced).
#pragma unroll
    for (int k = 0; k < 4; ++k) {
        const int p = p0 + ty + 8 * k;
        const int e = e0 + tx;
        if (p < PDIM) {
            pet[(size_t)p * EMB + e] = tile[tx][ty + 8 * k];
        }
    }
}

// ---------------------------------------------------------------------------
// Kernel 2a: gather from transposed table (contiguous 6KB rows) + bias.
// grid = NSAMP/SPB blocks, block = 384 threads (12 waves), v4f per thread.
// Output is write-once -> non-temporal stores keep L2 for pet row reuse.
// ---------------------------------------------------------------------------
__global__ __launch_bounds__(ROW4) void gather_bias_T_kernel(
        const float* __restrict__ x1, const float* __restrict__ x2,
        const float* __restrict__ pet, const float* __restrict__ bias,
        float* __restrict__ out) {
    __shared__ int sp[SPB];

    const int t    = threadIdx.x;          // 0..383 -> v4f slot in row
    const int base = blockIdx.x * SPB;

    if (t < SPB) {
        const int n = base + t;
        sp[t] = bin_index(x1[n]) * BINS + bin_index(x2[n]);
    }

    const v4f b4 = ((const v4f*)bias)[t];
    __syncthreads();

    const bool pf_lane = ((t & 7) == 0);   // one prefetch per 128B line

#pragma unroll 4
    for (int i = 0; i < SPB; ++i) {
        if (pf_lane && (i + 2 < SPB)) {
            // gfx1250: lowers to global_prefetch_b8; 2 rows ahead
            __builtin_prefetch(pet + (size_t)sp[i + 2] * EMB + (size_t)t * 4, 0, 1);
        }
        const v4f v = ((const v4f*)(pet + (size_t)sp[i] * EMB))[t];
        const v4f r = v + b4;
        __builtin_nontemporal_store(r, (v4f*)(out + (size_t)(base + i) * EMB) + t);
    }
}

// ---------------------------------------------------------------------------
// Kernel 2b: fallback direct gather from native [EMB][257][257] layout
// (used only if the workspace cannot hold the transposed table).
// ---------------------------------------------------------------------------
__global__ __launch_bounds__(ROW4) void gather_bias_direct_kernel(
        const float* __restrict__ x1, const float* __restrict__ x2,
        const float* __restrict__ pe, const float* __restrict__ bias,
        float* __restrict__ out) {
    __shared__ int sp[SPB];

    const int t    = threadIdx.x;
    const int base = blockIdx.x * SPB;

    if (t < SPB) {
        const int n = base + t;
        sp[t] = bin_index(x1[n]) * BINS + bin_index(x2[n]);
    }

    const v4f b4 = ((const v4f*)bias)[t];
    __syncthreads();

    const int e0 = t * 4;
    for (int i = 0; i < SPB; ++i) {
        const size_t off = (size_t)sp[i];
        v4f r;
        r.x = pe[(size_t)(e0 + 0) * PDIM + off] + b4.x;
        r.y = pe[(size_t)(e0 + 1) * PDIM + off] + b4.y;
        r.z = pe[(size_t)(e0 + 2) * PDIM + off] + b4.z;
        r.w = pe[(size_t)(e0 + 3) * PDIM + off] + b4.w;
        __builtin_nontemporal_store(r, (v4f*)(out + (size_t)(base + i) * EMB) + t);
    }
}

// ---------------------------------------------------------------------------
extern "C" void kernel_launch(void* const* d_in, const int* in_sizes, int n_in,
                              void* d_out, int out_size, void* d_ws, size_t ws_size,
                              hipStream_t stream) {
    const float* x1   = (const float*)d_in[0];
    const float* x2   = (const float*)d_in[1];
    const float* pe   = (const float*)d_in[2];   // [EMB, 257, 257]
    const float* bias = (const float*)d_in[3];   // [EMB]
    float*       out  = (float*)d_out;           // [NSAMP, EMB]

    const size_t pet_bytes = (size_t)PDIM * (size_t)EMB * sizeof(float);

    if (ws_size >= pet_bytes) {
        float* pet = (float*)d_ws;
        dim3 tgrid((PDIM + 31) / 32, EMB / 32);  // (2065, 48)
        dim3 tblk(32, 8);
        pe_transpose_kernel<<<tgrid, tblk, 0, stream>>>(pe, pet);
        gather_bias_T_kernel<<<NSAMP / SPB, ROW4, 0, stream>>>(x1, x2, pet, bias, out);
    } else {
        gather_bias_direct_kernel<<<NSAMP / SPB, ROW4, 0, stream>>>(x1, x2, pe, bias, out);
    }
}